// HaarIDWT_86474871537996
// MI455X (gfx1250) — compile-verified
//
#include <hip/hip_runtime.h>

typedef __attribute__((ext_vector_type(2))) float v2f;
typedef __attribute__((ext_vector_type(8))) float v8f;

#define IN_W   256
#define OUT_W  512
#define WAVES_PER_BLOCK 8

// One wave32 processes one input row (one (b,c,h) line of 256 pixels) as
// 16 WMMA tiles of 16 pixels each, phase-separated in groups of 4 tiles so
// 8 NT loads are in flight before the 4 WMMAs and 8 NT stores:
//   D(16x16) = A(16x4, constant Haar synthesis rows x0.5) * B(4x16, pixel data)
// B layout (32-bit B-matrix, K striped across lane halves):
//   VGPR0: lanes 0-15 = K=0 (ll), lanes 16-31 = K=2 (hl)
//   VGPR1: lanes 0-15 = K=1 (lh), lanes 16-31 = K=3 (hh)
// D layout: VGPR v, lanes 0-15 = row M=v, N=lane. So lane n holds
//   c[0]=tl, c[1]=tr, c[2]=bl, c[3]=br of pixel n -> two b64 NT stores.
__global__ __launch_bounds__(256) void haar_idwt_wmma_kernel(
    const float* __restrict__ ll, const float* __restrict__ lh,
    const float* __restrict__ hl, const float* __restrict__ hh,
    float* __restrict__ out, unsigned total_rows)
{
    const unsigned tid  = threadIdx.x;
    const unsigned lane = tid & 31u;
    const unsigned wave = tid >> 5;
    const unsigned row  = blockIdx.x * WAVES_PER_BLOCK + wave;  // = bc*256 + h
    if (row >= total_rows) return;           // wave-uniform: EXEC stays all-1s

    const unsigned bc = row >> 8;            // b*64 + c
    const unsigned h  = row & 255u;

    const size_t in_base  = (size_t)row * IN_W;                      // bc*256*256 + h*256
    const size_t out_base = (size_t)bc * (OUT_W * OUT_W) + (size_t)(2u * h) * OUT_W;

    const unsigned m  = lane & 15u;          // pixel index within tile / A-matrix row
    const bool     hi = lane >= 16u;         // upper half-wave: K=2,3 operands

    // Per-lane source streams for the B matrix.
    const float* __restrict__ s0 = hi ? hl : ll;   // K=2 : K=0
    const float* __restrict__ s1 = hi ? hh : lh;   // K=3 : K=1

    // A-matrix: rows 0-3 are the Haar synthesis rows (tl,tr,bl,br), rows 4-15 zero.
    //   tl = .5*(ll - lh - hl + hh)   r=0 (vr=0,hr=0)
    //   tr = .5*(ll - lh + hl - hh)   r=1 (vr=0,hr=1)
    //   bl = .5*(ll + lh - hl - hh)   r=2 (vr=1,hr=0)
    //   br = .5*(ll + lh + hl + hh)   r=3 (vr=1,hr=1)
    // Lane layout: lane m holds A[m][K0],A[m][K1] with (K0,K1)=(0,1) lo half, (2,3) hi half.
    float a0 = 0.0f, a1 = 0.0f;
    if (m < 4u) {
        const unsigned vr = m >> 1, hr = m & 1u;
        const float sK1 = vr ? 0.5f : -0.5f;            // coeff of lh
        const float sK2 = hr ? 0.5f : -0.5f;            // coeff of hl
        const float sK3 = (vr ^ hr) ? -0.5f : 0.5f;     // coeff of hh
        a0 = hi ? sK2 : 0.5f;                           // K=2 : K=0 (ll coeff = +.5)
        a1 = hi ? sK3 : sK1;                            // K=3 : K=1
    }
    v2f A; A[0] = a0; A[1] = a1;

    const size_t col_lo = 2u * m;            // output column pair for this lane

    for (unsigned w0 = 0; w0 < IN_W; w0 += 64u) {      // 4 groups of 4 tiles
        const size_t off = in_base + w0 + m;

        // Streaming prefetch of the next group (speculative; OOB is dropped).
        __builtin_prefetch(s0 + off + 64u, 0, 1);
        __builtin_prefetch(s1 + off + 64u, 0, 1);

        // ---- Phase 1: 8 NT loads in flight ----
        v2f B0, B1, B2, B3;
        B0[0] = __builtin_nontemporal_load(s0 + off);
        B0[1] = __builtin_nontemporal_load(s1 + off);
        B1[0] = __builtin_nontemporal_load(s0 + off + 16u);
        B1[1] = __builtin_nontemporal_load(s1 + off + 16u);
        B2[0] = __builtin_nontemporal_load(s0 + off + 32u);
        B2[1] = __builtin_nontemporal_load(s1 + off + 32u);
        B3[0] = __builtin_nontemporal_load(s0 + off + 48u);
        B3[1] = __builtin_nontemporal_load(s1 + off + 48u);

        // ---- Phase 2: 4 independent WMMAs ----
        const v8f z = {0.f, 0.f, 0.f, 0.f, 0.f, 0.f, 0.f, 0.f};
        v8f c0 = __builtin_amdgcn_wmma_f32_16x16x4_f32(false, A, false, B0, (short)0, z, false, false);
        v8f c1 = __builtin_amdgcn_wmma_f32_16x16x4_f32(false, A, false, B1, (short)0, z, false, false);
        v8f c2 = __builtin_amdgcn_wmma_f32_16x16x4_f32(false, A, false, B2, (short)0, z, false, false);
        v8f c3 = __builtin_amdgcn_wmma_f32_16x16x4_f32(false, A, false, B3, (short)0, z, false, false);

        // ---- Phase 3: 8 NT stores (lanes 0-15 hold the pixel results) ----
        if (!hi) {  // reconverges before the next group's WMMAs
            const size_t col = 2u * w0 + col_lo;
            float* r0 = out + out_base + col;            // output row 2h
            float* r1 = r0 + OUT_W;                      // output row 2h+1
            v2f t, b;
            t[0] = c0[0]; t[1] = c0[1]; b[0] = c0[2]; b[1] = c0[3];
            __builtin_nontemporal_store(t, reinterpret_cast<v2f*>(r0));
            __builtin_nontemporal_store(b, reinterpret_cast<v2f*>(r1));
            t[0] = c1[0]; t[1] = c1[1]; b[0] = c1[2]; b[1] = c1[3];
            __builtin_nontemporal_store(t, reinterpret_cast<v2f*>(r0 + 32));
            __builtin_nontemporal_store(b, reinterpret_cast<v2f*>(r1 + 32));
            t[0] = c2[0]; t[1] = c2[1]; b[0] = c2[2]; b[1] = c2[3];
            __builtin_nontemporal_store(t, reinterpret_cast<v2f*>(r0 + 64));
            __builtin_nontemporal_store(b, reinterpret_cast<v2f*>(r1 + 64));
            t[0] = c3[0]; t[1] = c3[1]; b[0] = c3[2]; b[1] = c3[3];
            __builtin_nontemporal_store(t, reinterpret_cast<v2f*>(r0 + 96));
            __builtin_nontemporal_store(b, reinterpret_cast<v2f*>(r1 + 96));
        }
    }
}

extern "C" void kernel_launch(void* const* d_in, const int* in_sizes, int n_in,
                              void* d_out, int out_size, void* d_ws, size_t ws_size,
                              hipStream_t stream) {
    const float* ll = (const float*)d_in[0];
    const float* lh = (const float*)d_in[1];
    const float* hl = (const float*)d_in[2];
    const float* hh = (const float*)d_in[3];
    float* out = (float*)d_out;

    // in_sizes[0] = 8*64*256*256; one wave per input row of 256 pixels.
    const unsigned total_rows = (unsigned)(in_sizes[0] / IN_W);          // 131072
    const unsigned blocks = (total_rows + WAVES_PER_BLOCK - 1) / WAVES_PER_BLOCK;  // 16384

    haar_idwt_wmma_kernel<<<blocks, 256, 0, stream>>>(ll, lh, hl, hh, out, total_rows);
}